// AttnBlock_49538152792944
// MI455X (gfx1250) — compile-verified
//
#include <hip/hip_runtime.h>

// ---------------------------------------------------------------------------
// Equivariant attention block for MI455X (gfx1250, wave32).
// Radial-MLP matmuls run on V_WMMA_F32_16X16X4_F32 (f32 matrix pipe);
// batch-norms are folded into affine/weight transforms by prep kernels so the
// hot kernel is BN-free. wj streaming (~1.04 GB) is the roofline floor
// (~45 us at 23.3 TB/s); it is prefetched during the WMMA phase and read
// non-temporally. ReLU = single v_med3_f32.
// ---------------------------------------------------------------------------

typedef float v2f __attribute__((ext_vector_type(2)));
typedef float v8f __attribute__((ext_vector_type(8)));

#define EPSBN 1e-5f

// per p = l*3+k: J = 2*min(l,k)+1, r_size = 4*J, prefix offsets, #M-tiles in layer3
constexpr int RSZ9[9]   = {4, 4, 4, 4, 12, 12, 4, 12, 20};
constexpr int RBASE9[9] = {0, 4, 8, 12, 16, 28, 40, 44, 56};   // sum = 76
constexpr int MT9[9]    = {1, 1, 1, 1, 1, 1, 1, 1, 2};

// single-instruction ReLU: v_med3_f32(x, 0, FLT_MAX) == clamp(x, 0, MAX)
__device__ inline float relu1(float x) {
  return __builtin_amdgcn_fmed3f(x, 0.0f, 3.402823466e+38f);
}

__device__ inline float wave_sum32(float v) {
#pragma unroll
  for (int m = 16; m > 0; m >>= 1) v += __shfl_xor(v, m, 32);
  return v;
}

// ---------------- Stage 1: mean/var of dist ----------------
__global__ __launch_bounds__(256) void k_dstats(const float* __restrict__ dist, int E,
                                                float* __restrict__ st) {
  float s = 0.f, q = 0.f;
  for (int e = blockIdx.x * blockDim.x + threadIdx.x; e < E; e += gridDim.x * blockDim.x) {
    float d = dist[e];
    s += d;
    q = fmaf(d, d, q);
  }
  s = wave_sum32(s);
  q = wave_sum32(q);
  if ((threadIdx.x & 31) == 0) {
    atomicAdd(&st[0], s);
    atomicAdd(&st[1], q);
  }
}

// ---------------- Stage 2: fold BN1 into h1 = relu(A1*d + C1) ----------------
__global__ void k_fold1(const float* __restrict__ rW1, const float* __restrict__ rg1,
                        const float* __restrict__ rbb1, const float* __restrict__ st,
                        float Einv, float* __restrict__ A1, float* __restrict__ C1) {
  int t = threadIdx.x;
  if (t >= 90) return;  // P*10
  float mu  = st[0] * Einv;
  float var = fmaxf(st[1] * Einv - mu * mu, 0.f);
  float a = rW1[t];               // (P,10,1)
  float g = rg1[t], bb = rbb1[t];
  float rs = rsqrtf(a * a * var + EPSBN);
  float A = a * g * rs;           // b1 cancels under BN centering
  A1[t] = A;
  C1[t] = bb - A * mu;
}

// ---------------- Stage 3: mean/var of h2_pre (270 features) ----------------
__global__ __launch_bounds__(256) void k_h2stats(const float* __restrict__ dist, int E,
                                                 const float* __restrict__ A1,
                                                 const float* __restrict__ C1,
                                                 const float* __restrict__ rW2,
                                                 const float* __restrict__ rb2,
                                                 float* __restrict__ st2) {
  int p = blockIdx.y;
  const float* A = A1 + p * 10;
  const float* Cc = C1 + p * 10;
  const float* W = rW2 + p * 300;
  const float* b = rb2 + p * 30;
  float sum[30], sq[30];
#pragma unroll
  for (int m = 0; m < 30; ++m) { sum[m] = 0.f; sq[m] = 0.f; }
  for (int e = blockIdx.x * blockDim.x + threadIdx.x; e < E; e += gridDim.x * blockDim.x) {
    float d = dist[e];
    float h1[10];
#pragma unroll
    for (int j = 0; j < 10; ++j) h1[j] = relu1(fmaf(A[j], d, Cc[j]));
#pragma unroll
    for (int m = 0; m < 30; ++m) {
      float v = b[m];
#pragma unroll
      for (int j = 0; j < 10; ++j) v = fmaf(W[m * 10 + j], h1[j], v);
      sum[m] += v;
      sq[m] = fmaf(v, v, sq[m]);
    }
  }
#pragma unroll
  for (int m = 0; m < 30; ++m) {
    float a0 = wave_sum32(sum[m]);
    float a1 = wave_sum32(sq[m]);
    if ((threadIdx.x & 31) == 0) {
      atomicAdd(&st2[p * 30 + m], a0);
      atomicAdd(&st2[270 + p * 30 + m], a1);
    }
  }
}

// ---------------- Stage 4: fold BN2 into W2/b2; zero-pad W2(32x12), W3(32x32) ----------------
__global__ __launch_bounds__(1024) void k_fold2(
    const float* __restrict__ rW2, const float* __restrict__ rb2,
    const float* __restrict__ rg2, const float* __restrict__ rbb2,
    const float* __restrict__ rW3, const float* __restrict__ rb3,
    const float* __restrict__ st2, float Einv,
    float* __restrict__ W2F, float* __restrict__ B2F,
    float* __restrict__ W3P, float* __restrict__ B3P) {
  int p = blockIdx.x;
  int row = threadIdx.x >> 5, col = threadIdx.x & 31;
  int rsz = RSZ9[p];
  float s2 = 0.f, bias2 = 0.f;
  if (row < 30) {
    float mu  = st2[p * 30 + row] * Einv;
    float var = fmaxf(st2[270 + p * 30 + row] * Einv - mu * mu, 0.f);
    float rs = rsqrtf(var + EPSBN);
    s2 = rg2[p * 30 + row] * rs;
    bias2 = s2 * rb2[p * 30 + row] + rbb2[p * 30 + row] - s2 * mu;
  }
  if (col < 12)
    W2F[p * 384 + row * 12 + col] = (row < 30 && col < 10) ? s2 * rW2[p * 300 + row * 10 + col] : 0.f;
  if (col == 0) B2F[p * 32 + row] = (row < 30) ? bias2 : 0.f;
  W3P[p * 1024 + row * 32 + col] = (row < rsz && col < 30) ? rW3[p * 600 + row * 30 + col] : 0.f;
  if (col == 1) B3P[p * 32 + row] = (row < rsz) ? rb3[p * 20 + row] : 0.f;
}

// ---------------- epilogue einsum helper ----------------
template <int L, int K>
__device__ inline void pair_accum(const float* __restrict__ w, const float* __restrict__ R,
                                  const float (&fk)[2][5], float (&acc)[2][5]) {
  constexpr int J  = 2 * ((L < K) ? L : K) + 1;
  constexpr int DL = 2 * L + 1, DK = 2 * K + 1;
#pragma unroll
  for (int j = 0; j < J; ++j) {
    float r00 = R[j * 4 + 0], r01 = R[j * 4 + 1];
    float r10 = R[j * 4 + 2], r11 = R[j * 4 + 3];
#pragma unroll
    for (int lp = 0; lp < DL; ++lp) {
      float t0 = 0.f, t1 = 0.f;
#pragma unroll
      for (int kp = 0; kp < DK; ++kp) {
        float wv = __builtin_nontemporal_load(&w[(j * DL + lp) * DK + kp]);  // stream-once
        t0 = fmaf(wv, fk[0][kp], t0);
        t1 = fmaf(wv, fk[1][kp], t1);
      }
      acc[0][lp] = fmaf(r00, t0, fmaf(r01, t1, acc[0][lp]));
      acc[1][lp] = fmaf(r10, t0, fmaf(r11, t1, acc[1][lp]));
    }
  }
}

// ---------------- Stage 5: main kernel (WMMA MLP + einsum + exp + segment-sum) ----------------
// Block = 64 threads = 2 waves; each wave owns 32 edges (two 16-edge WMMA groups).
// LDS per wave: h2 tile 32x17 (544) + R transpose tile 32x81 (2592) = 3136 floats.
__global__ __launch_bounds__(64) void k_main(
    const int* __restrict__ dst, const float* __restrict__ dist,
    const float* __restrict__ f0, const float* __restrict__ f1, const float* __restrict__ f2,
    const float* __restrict__ wq,
    const float* __restrict__ wj00, const float* __restrict__ wj01, const float* __restrict__ wj02,
    const float* __restrict__ wj10, const float* __restrict__ wj11, const float* __restrict__ wj12,
    const float* __restrict__ wj20, const float* __restrict__ wj21, const float* __restrict__ wj22,
    const float* __restrict__ A1, const float* __restrict__ C1,
    const float* __restrict__ W2F, const float* __restrict__ B2F,
    const float* __restrict__ W3P, const float* __restrict__ B3P,
    float* __restrict__ eout, float* __restrict__ sseg, int E) {
  __shared__ float smem[2 * 3136];
  const int lane = threadIdx.x & 31;
  const int wid  = threadIdx.x >> 5;
  float* h2t = smem + wid * 3136;   // [32 feat rows][stride 17]
  float* RL  = h2t + 544;          // [32 edges][stride 81]
  const int n = lane & 15;
  const int half = lane >> 4;
  const long wave = (long)blockIdx.x * 2 + wid;

  // -------- resolve epilogue addresses early and prefetch the streams --------
  const long e  = wave * 32 + lane;
  const bool ok = e < E;
  const long ei = ok ? e : (long)E - 1;
  const int idx = dst[ei];
  // per-edge wj blocks: stream-once -> pull toward L2/L0 while WMMA phase runs
  __builtin_prefetch(wj22 + (size_t)ei * 125, 0, 1);
  __builtin_prefetch(wj22 + (size_t)ei * 125 + 32, 0, 1);
  __builtin_prefetch(wj22 + (size_t)ei * 125 + 64, 0, 1);
  __builtin_prefetch(wj22 + (size_t)ei * 125 + 96, 0, 1);
  __builtin_prefetch(wj21 + (size_t)ei * 45, 0, 1);
  __builtin_prefetch(wj21 + (size_t)ei * 45 + 32, 0, 1);
  __builtin_prefetch(wj12 + (size_t)ei * 45, 0, 1);
  __builtin_prefetch(wj12 + (size_t)ei * 45 + 32, 0, 1);
  __builtin_prefetch(wj11 + (size_t)ei * 27, 0, 1);
  __builtin_prefetch(wj02 + (size_t)ei * 5, 0, 1);
  __builtin_prefetch(wj20 + (size_t)ei * 5, 0, 1);
  __builtin_prefetch(wj10 + (size_t)ei * 3, 0, 1);
  __builtin_prefetch(wj01 + (size_t)ei * 3, 0, 1);
  __builtin_prefetch(wj00 + (size_t)ei, 0, 1);
  __builtin_prefetch(f1 + idx * 6, 0, 1);
  __builtin_prefetch(f2 + idx * 10, 0, 1);

#pragma unroll
  for (int g = 0; g < 2; ++g) {
    long eb = wave * 32 + g * 16 + n;
    long eg = eb < E ? eb : (long)E - 1;
    float d = dist[eg];
#pragma unroll
    for (int p = 0; p < 9; ++p) {
      const float* A1p = A1 + p * 10;
      const float* C1p = C1 + p * 10;
      // Each lane needs only h1 features j in {2h, 2h+1, 4+2h, 5+2h, 8, 9};
      // the (s=2, half=1) pair is the K-padding -> zero. Address-based
      // indexing keeps register indices compile-time (no v_cndmask trees).
      float h1v[6];
      {
        int j0 = 2 * half, j1 = 4 + 2 * half;
        h1v[0] = relu1(fmaf(A1p[j0],     d, C1p[j0]));
        h1v[1] = relu1(fmaf(A1p[j0 + 1], d, C1p[j0 + 1]));
        h1v[2] = relu1(fmaf(A1p[j1],     d, C1p[j1]));
        h1v[3] = relu1(fmaf(A1p[j1 + 1], d, C1p[j1 + 1]));
        float v8 = relu1(fmaf(A1p[8], d, C1p[8]));
        float v9 = relu1(fmaf(A1p[9], d, C1p[9]));
        h1v[4] = half ? 0.f : v8;
        h1v[5] = half ? 0.f : v9;
      }
      // layer 2: h2^T(30x16) = W2f(30x12-pad) @ h1^T(12x16), 2 M-tiles x 3 K-steps
#pragma unroll
      for (int t = 0; t < 2; ++t) {
        v8f c;
#pragma unroll
        for (int r = 0; r < 8; ++r) c[r] = B2F[p * 32 + t * 16 + r + 8 * half];
#pragma unroll
        for (int s = 0; s < 3; ++s) {
          const float* ar = W2F + p * 384 + (t * 16 + n) * 12 + 4 * s + 2 * half;
          v2f a; a.x = ar[0]; a.y = ar[1];                 // A: halves split K
          v2f b; b.x = h1v[2 * s]; b.y = h1v[2 * s + 1];   // B: halves split K
          c = __builtin_amdgcn_wmma_f32_16x16x4_f32(false, a, false, b, (short)0, c, false, false);
        }
#pragma unroll
        for (int r = 0; r < 8; ++r)
          h2t[(t * 16 + r + 8 * half) * 17 + n] = relu1(c[r]);  // ReLU + transpose stage
      }
      // layer 3: R^T(rsz x 16) = W3p(32x32-pad) @ h2^T(32x16), 8 K-steps
#pragma unroll
      for (int t3 = 0; t3 < 2; ++t3) {
        if (t3 < MT9[p]) {
          v8f c;
#pragma unroll
          for (int r = 0; r < 8; ++r) c[r] = B3P[p * 32 + t3 * 16 + r + 8 * half];
#pragma unroll
          for (int s = 0; s < 8; ++s) {
            const float* ar = W3P + p * 1024 + (t3 * 16 + n) * 32 + 4 * s + 2 * half;
            v2f a; a.x = ar[0]; a.y = ar[1];
            v2f b;
            b.x = h2t[(4 * s + 2 * half) * 17 + n];
            b.y = h2t[(4 * s + 2 * half + 1) * 17 + n];
            c = __builtin_amdgcn_wmma_f32_16x16x4_f32(false, a, false, b, (short)0, c, false, false);
          }
#pragma unroll
          for (int r = 0; r < 8; ++r) {
            int row = t3 * 16 + r + 8 * half;
            if (row < RSZ9[p])  // compile-time after unroll; EXEC stays full for WMMAs
              RL[(g * 16 + n) * 81 + RBASE9[p] + row] = c[r];
          }
        }
      }
    }
  }

  // -------- epilogue: one edge per lane (32 edges/wave) --------
  float fa[3][2][5];
#pragma unroll
  for (int c2 = 0; c2 < 2; ++c2) {
    fa[0][c2][0] = f0[idx * 2 + c2];
#pragma unroll
    for (int m = 0; m < 3; ++m) fa[1][c2][m] = f1[idx * 6 + c2 * 3 + m];
#pragma unroll
    for (int m = 0; m < 5; ++m) fa[2][c2][m] = f2[idx * 10 + c2 * 5 + m];
  }
  float acc[3][2][5];
#pragma unroll
  for (int l = 0; l < 3; ++l)
#pragma unroll
    for (int o = 0; o < 2; ++o)
#pragma unroll
      for (int m = 0; m < 5; ++m) acc[l][o][m] = 0.f;

  const float* RLl = RL + lane * 81;  // stride 81 (odd): conflict-free across 32 lanes
  const float* wjp[9] = {wj00, wj01, wj02, wj10, wj11, wj12, wj20, wj21, wj22};

#define PAIR(L, K)                                                                         \
  do {                                                                                     \
    constexpr int Jc = 2 * (((L) < (K)) ? (L) : (K)) + 1;                                  \
    constexpr int CNT = Jc * (2 * (L) + 1) * (2 * (K) + 1);                                \
    pair_accum<L, K>(wjp[(K)*3 + (L)] + (size_t)ei * CNT, RLl + RBASE9[(L)*3 + (K)],       \
                     fa[K], acc[L]);                                                       \
  } while (0)
  PAIR(0, 0); PAIR(0, 1); PAIR(0, 2);
  PAIR(1, 0); PAIR(1, 1); PAIR(1, 2);
  PAIR(2, 0); PAIR(2, 1); PAIR(2, 2);
#undef PAIR

  float wql[12];
#pragma unroll
  for (int i = 0; i < 12; ++i) wql[i] = wq[i];
  float dot = 0.f;
#pragma unroll
  for (int l = 0; l < 3; ++l)
#pragma unroll
    for (int o = 0; o < 2; ++o)
#pragma unroll
      for (int m = 0; m < 5; ++m)
        if (m < 2 * l + 1) {
          float qv = fmaf(wql[l * 4 + o * 2 + 0], fa[l][0][m],
                          wql[l * 4 + o * 2 + 1] * fa[l][1][m]);
          dot = fmaf(qv, acc[l][o][m], dot);
        }
  float ev = expf(dot);
  if (ok) {
    eout[e] = ev;
    atomicAdd(&sseg[idx], ev);
  }
}

// ---------------- Stage 6: normalize ----------------
__global__ __launch_bounds__(256) void k_norm(const float* __restrict__ eo,
                                              const int* __restrict__ dst,
                                              const float* __restrict__ sseg,
                                              float* __restrict__ out, int E) {
  int e = blockIdx.x * blockDim.x + threadIdx.x;
  if (e < E) out[e] = eo[e] / sseg[dst[e]];
}

// ---------------------------------------------------------------------------
extern "C" void kernel_launch(void* const* d_in, const int* in_sizes, int n_in,
                              void* d_out, int out_size, void* d_ws, size_t ws_size,
                              hipStream_t stream) {
  (void)n_in; (void)out_size; (void)ws_size;
  const int*   dst  = (const int*)d_in[1];
  const float* dist = (const float*)d_in[2];
  const float* f0 = (const float*)d_in[3];
  const float* f1 = (const float*)d_in[4];
  const float* f2 = (const float*)d_in[5];
  const float* wq = (const float*)d_in[6];
  const float* wj[9];
  for (int i = 0; i < 9; ++i) wj[i] = (const float*)d_in[7 + i];
  const float* rW1  = (const float*)d_in[16];
  const float* rg1  = (const float*)d_in[18];
  const float* rbb1 = (const float*)d_in[19];
  const float* rW2  = (const float*)d_in[20];
  const float* rb2  = (const float*)d_in[21];
  const float* rg2  = (const float*)d_in[22];
  const float* rbb2 = (const float*)d_in[23];
  const float* rW3  = (const float*)d_in[24];
  const float* rb3  = (const float*)d_in[25];

  const int E = in_sizes[2];
  const int N = in_sizes[3] / 2;  // f0 has N*C*1 elements
  float* wsf = (float*)d_ws;

  // workspace layout (floats)
  size_t oE   = 0;
  size_t oS   = oE + (size_t)E;      // segment sums [N]
  size_t oDST = oS + (size_t)N;      // dist stats [2]
  size_t oH2  = oDST + 2;            // h2 stats   [540]
  size_t oA1  = oH2 + 540;           // [90]
  size_t oC1  = oA1 + 90;            // [90]
  size_t oW2F = oC1 + 90;            // [9*32*12]
  size_t oB2F = oW2F + 3456;         // [9*32]
  size_t oW3P = oB2F + 288;          // [9*32*32]
  size_t oB3P = oW3P + 9216;         // [9*32]

  hipMemsetAsync(wsf + oS, 0, (size_t)(N + 2 + 540) * sizeof(float), stream);
  const float Einv = 1.0f / (float)E;

  k_dstats<<<256, 256, 0, stream>>>(dist, E, wsf + oDST);
  k_fold1<<<1, 128, 0, stream>>>(rW1, rg1, rbb1, wsf + oDST, Einv, wsf + oA1, wsf + oC1);
  k_h2stats<<<dim3(64, 9), 256, 0, stream>>>(dist, E, wsf + oA1, wsf + oC1, rW2, rb2, wsf + oH2);
  k_fold2<<<9, 1024, 0, stream>>>(rW2, rb2, rg2, rbb2, rW3, rb3, wsf + oH2, Einv,
                                  wsf + oW2F, wsf + oB2F, wsf + oW3P, wsf + oB3P);
  int nb = (E + 63) / 64;  // 64 threads = 2 waves = 64 edges per block
  k_main<<<nb, 64, 0, stream>>>(dst, dist, f0, f1, f2, wq,
                                wj[0], wj[1], wj[2], wj[3], wj[4], wj[5], wj[6], wj[7], wj[8],
                                wsf + oA1, wsf + oC1, wsf + oW2F, wsf + oB2F,
                                wsf + oW3P, wsf + oB3P, wsf + oE, wsf + oS, E);
  k_norm<<<(E + 255) / 256, 256, 0, stream>>>(wsf + oE, dst, wsf + oS, (float*)d_out, E);
}